// RecoverLayer_39333310497211
// MI455X (gfx1250) — compile-verified
//
#include <hip/hip_runtime.h>

#define HH 480
#define WW 640
#define BB 4
#define NPIX (BB * HH * WW)
#define W_FRAC (31.0f / 32.0f)

typedef unsigned int tdm_u32x4 __attribute__((ext_vector_type(4)));
typedef int          tdm_i32x4 __attribute__((ext_vector_type(4)));
typedef int          tdm_i32x8 __attribute__((ext_vector_type(8)));

// ---------------------------------------------------------------------------
// Kernel 1: dark channel  d(b,y,x) = min_c hazeImg(b,y,x,c)   -> level 0
// (compiles to global_load_b96 + v_min3_num_f32: pure streaming, HBM-bound)
// ---------------------------------------------------------------------------
__global__ void dark_channel_kernel(const float* __restrict__ haze,
                                    float* __restrict__ lvl0) {
    int idx = blockIdx.x * blockDim.x + threadIdx.x;
    if (idx >= NPIX) return;
    const float* p = haze + 3 * (size_t)idx;
    float a = p[0], b = p[1], c = p[2];
    lvl0[idx] = fminf(a, fminf(b, c));
}

// ---------------------------------------------------------------------------
// Kernel 2: pyramid build  T_{j+1}[y][x] = min of T_j at (y,x),(y+s,x),(y,x+s),(y+s,x+s)
// The (16+s) x (64+s) input halo tile is staged into LDS with ONE Tensor Data
// Mover descriptor (tensor_load_to_lds), then each output is 4 LDS reads +
// 3 v_min_num_f32. Halo elements past the image edge feed only pyramid
// entries at y > H-2^(j+1) / x > W-2^(j+1), which valid queries never read,
// so the OOB fill value is a don't-care.
// ---------------------------------------------------------------------------
#define TH 16
#define TW 64
__global__ void build_level_kernel(const float* __restrict__ src,
                                   float* __restrict__ dst, int s) {
    __shared__ float lds[(TH + 32) * (TW + 32)];   // dense tile, pitch = IW
    const int b   = blockIdx.z;
    const int ty0 = blockIdx.y * TH;
    const int tx0 = blockIdx.x * TW;
    const int IH  = TH + s;
    const int IW  = TW + s;
    const float* sp = src + (size_t)b * HH * WW;

    if (threadIdx.x < 32) {   // first wave issues the TDM descriptor
        unsigned long long gaddr = (unsigned long long)(sp + (size_t)ty0 * WW + tx0);
        // low 32 bits of a flat LDS pointer == LDS byte offset (ISA 10.2)
        unsigned lds_base = (unsigned)(unsigned long long)&lds[0];
        unsigned tdim0 = (unsigned)(WW - tx0);   // cols remaining from tile start (OOB rel. to tile)
        unsigned tdim1 = (unsigned)(HH - ty0);   // rows remaining from tile start

        tdm_u32x4 g0;
        g0.x = 1u;                                    // count=1, user descriptor
        g0.y = lds_base;                              // lds_addr [63:32]
        g0.z = (unsigned)(gaddr & 0xFFFFFFFFu);       // global_addr[31:0]
        g0.w = (unsigned)((gaddr >> 32) & 0x1FFFFFFu) // global_addr[56:32]
             | (2u << 30);                            // type = 2 ("image")

        tdm_i32x8 g1;
        g1[0] = (int)(2u << 16);                                   // data_size = 4B
        g1[1] = (int)((tdim0 & 0xFFFFu) << 16);                    // tensor_dim0[15:0]
        g1[2] = (int)((tdim0 >> 16) | ((tdim1 & 0xFFFFu) << 16));  // dim0 hi | dim1 lo
        g1[3] = (int)((tdim1 >> 16) | ((unsigned)IW << 16));       // dim1 hi | tile_dim0
        g1[4] = (int)(unsigned)IH;                                 // tile_dim1 (tile_dim2=0)
        g1[5] = (int)(unsigned)WW;                                 // tensor_dim0_stride[31:0]
        g1[6] = 0;                                                 // stride hi | dim1_stride lo
        g1[7] = 0;

        tdm_i32x4 gz = {0, 0, 0, 0};
#if __clang_major__ >= 23
        tdm_i32x8 gz8 = {0, 0, 0, 0, 0, 0, 0, 0};
        __builtin_amdgcn_tensor_load_to_lds(g0, g1, gz, gz, gz8, 0);
#else
        __builtin_amdgcn_tensor_load_to_lds(g0, g1, gz, gz, 0);
#endif
        __builtin_amdgcn_s_wait_tensorcnt((short)0);
    }
    __syncthreads();

    float* dp = dst + (size_t)b * HH * WW;
    const int srow = s * IW;
    for (int oe = threadIdx.x; oe < TH * TW; oe += blockDim.x) {
        int oy = oe >> 6;           // / TW
        int ox = oe & 63;           // % TW
        const float* l0 = lds + oy * IW + ox;
        float v00 = l0[0];
        float v01 = l0[s];
        float v10 = l0[srow];
        float v11 = l0[srow + s];
        dp[(ty0 + oy) * WW + (tx0 + ox)] = fminf(fminf(v00, v01), fminf(v10, v11));
    }
}

// ---------------------------------------------------------------------------
// Kernel 3: per-pixel O(1) windowed-min query + transmission map.
//   k  = max(1, int(patchMap));  rows [y-pl, y+ph] clipped (pl=(k-1)/2),
//   j  = min(floor(log2 Lr), floor(log2 Lc));  clipping keeps maxL <= 2*minL,
//   so <=4 x <=2 overlapping 2^j squares cover the window.
//   out = clip(1 - (31/32) * m / A[b], 0, 1)
// ---------------------------------------------------------------------------
__global__ void query_kernel(const float* __restrict__ patchMap,
                             const float* __restrict__ A,
                             const float* __restrict__ pyr,
                             float* __restrict__ out) {
    int idx = blockIdx.x * blockDim.x + threadIdx.x;
    if (idx >= NPIX) return;
    int b   = idx / (HH * WW);
    int rem = idx - b * (HH * WW);
    int y   = rem / WW;
    int x   = rem - y * WW;

    int k = (int)patchMap[idx];
    k = max(k, 1);
    int pl = (k - 1) >> 1;
    int ph = k - 1 - pl;

    int r0 = max(0, y - pl), r1 = min(HH - 1, y + ph);
    int c0 = max(0, x - pl), c1 = min(WW - 1, x + ph);
    int Lr = r1 - r0 + 1, Lc = c1 - c0 + 1;

    int jr = 31 - __clz(Lr);
    int jc = 31 - __clz(Lc);
    int j  = min(jr, jc);            // 0..6
    int s  = 1 << j;

    const float* T = pyr + (size_t)j * NPIX + (size_t)b * HH * WW;
    int rl = r1 - s + 1;
    int cl = c1 - s + 1;
    float m = 3.402823466e38f;
    for (int r = r0;; r += s) {
        int rr = min(r, rl);
        const float* Trow = T + rr * WW;
        for (int c = c0;; c += s) {
            int cc = min(c, cl);
            m = fminf(m, Trow[cc]);
            if (c >= cl) break;
        }
        if (r >= rl) break;
    }

    float v = 1.0f - W_FRAC * m / A[b];
    out[idx] = fminf(fmaxf(v, 0.0f), 1.0f);
}

// ---------------------------------------------------------------------------
extern "C" void kernel_launch(void* const* d_in, const int* in_sizes, int n_in,
                              void* d_out, int out_size, void* d_ws, size_t ws_size,
                              hipStream_t stream) {
    const float* patchMap = (const float*)d_in[0];   // (B,H,W,1)
    const float* hazeImg  = (const float*)d_in[1];   // (B,H,W,3)
    const float* A        = (const float*)d_in[2];   // (B,1)
    float* out = (float*)d_out;
    float* pyr = (float*)d_ws;                        // 7 levels * NPIX floats (34.4 MB)

    dark_channel_kernel<<<NPIX / 256, 256, 0, stream>>>(hazeImg, pyr);

    dim3 bgrid(WW / TW, HH / TH, BB);                 // 10 x 30 x 4
    for (int j = 0; j < 6; ++j) {
        build_level_kernel<<<bgrid, 256, 0, stream>>>(
            pyr + (size_t)j * NPIX, pyr + (size_t)(j + 1) * NPIX, 1 << j);
    }

    query_kernel<<<NPIX / 256, 256, 0, stream>>>(patchMap, A, pyr, out);
}